// EditDistStatModel_48369921688176
// MI455X (gfx1250) — compile-verified
//
#include <hip/hip_runtime.h>
#include <stdint.h>

#define TT 2048
#define VV 2048
#define NCC 10201
#define NEGINF (-__builtin_inff())

typedef __attribute__((ext_vector_type(2))) float v2f;
typedef __attribute__((ext_vector_type(8))) float v8f;

// ---------------- helpers ----------------

__device__ __forceinline__ float lse2(float a, float b) {
    float mx = fmaxf(a, b);
    if (mx == NEGINF) return NEGINF;
    float mn = fminf(a, b);
    return mx + __logf(1.0f + __expf(mn - mx));
}

// order-preserving float <-> uint encoding for atomicMax
__device__ __forceinline__ unsigned encF(float f) {
    unsigned u = __float_as_uint(f);
    return (u & 0x80000000u) ? ~u : (u | 0x80000000u);
}
__device__ __forceinline__ float decF(unsigned u) {
    return __uint_as_float((u & 0x80000000u) ? (u & 0x7fffffffu) : ~u);
}

// async global -> LDS staging (CDNA5 GLOBAL_LOAD_ASYNC_TO_LDS, GVS mode)
__device__ __forceinline__ void async_load_f32(unsigned ldsByteAddr, const float* base,
                                               unsigned byteOff) {
    asm volatile("global_load_async_to_lds_b32 %0, %1, %2 offset:0"
                 :: "v"(ldsByteAddr), "v"(byteOff), "s"(base)
                 : "memory");
}
__device__ __forceinline__ void wait_async0() {
    asm volatile("s_wait_asynccnt 0" ::: "memory");
}

// Block-wide inclusive scan over pairs (A,B) with the log-domain linear-recurrence
// operator: (A1,B1) o (A2,B2) = (A1+A2, lse(B1+A2, B2)). Returns exclusive prefix too.
// blockDim.x must be 1024 (32 waves of wave32). sA/sB are 32-entry LDS scratch.
__device__ __forceinline__ void block_scan_pair(float& A, float& B, float& Aex, float& Bex,
                                                float* sA, float* sB) {
    const int tid  = threadIdx.x;
    const int lane = tid & 31;
    const int wave = tid >> 5;
#pragma unroll
    for (int d = 1; d < 32; d <<= 1) {
        float Au = __shfl_up(A, d);
        float Bu = __shfl_up(B, d);
        if (lane >= d) { B = lse2(Bu + A, B); A = Au + A; }
    }
    if (lane == 31) { sA[wave] = A; sB[wave] = B; }
    __syncthreads();
    if (wave == 0) {
        float a = sA[lane];
        float b = sB[lane];
#pragma unroll
        for (int d = 1; d < 32; d <<= 1) {
            float au = __shfl_up(a, d);
            float bu = __shfl_up(b, d);
            if (lane >= d) { b = lse2(bu + a, b); a = au + a; }
        }
        sA[lane] = a; sB[lane] = b;
    }
    __syncthreads();
    float Aw = (wave == 0) ? 0.0f : sA[wave - 1];
    float Bw = (wave == 0) ? NEGINF : sB[wave - 1];
    float Ainc = Aw + A;
    float Binc = lse2(Bw + A, B);
    float Apx = __shfl_up(Ainc, 1);
    float Bpx = __shfl_up(Binc, 1);
    if (lane == 0) { Apx = Aw; Bpx = Bw; }
    Aex = Apx; Bex = Bpx;
    A = Ainc; B = Binc;
    __syncthreads();   // protect sA/sB for next call
}

// ---------------- K0: init ----------------
__global__ void k0_init(const int* ar, const int* en, const float* w,
                        float* roww, float* colw, float* sums, unsigned* mEnc) {
    int t = blockIdx.x * blockDim.x + threadIdx.x;
    if (t < TT) roww[t] = w[1 + ar[t]];
    if (t < VV) colw[t] = w[101 + en[t]];
    if (t < NCC) sums[t] = 0.0f;
    if (t == 0) *mEnc = encF(NEGINF);
}

// ---------------- K1: alpha (block 0) / beta (block 1) via row log-scan ----------------
__global__ void __launch_bounds__(1024)
k1_fwd_bwd(const int* ar, const int* en, const float* w,
           const float* roww, const float* colw,
           float* alpha, float* beta) {
    __shared__ float prevLds[VV + 1];
    __shared__ float wsubLds[100];
    __shared__ float sA[32], sB[32];

    const int rev = blockIdx.x;              // 0 = alpha (forward), 1 = beta (backward)
    float* mat = rev ? beta : alpha;
    const int t  = threadIdx.x;
    const int r0 = 2 * t, r1 = 2 * t + 1;    // scan coordinates
    const int j0 = rev ? (VV - 1 - r0) : r0; // mapped matrix columns
    const int j1 = rev ? (VV - 1 - r1) : r1;
    const float c0 = colw[j0];
    const float c1 = colw[j1];
    const int   e0 = en[j0];
    const int   e1 = en[j1];

    // ----- boundary row: prefix sums of mapped w_ins -----
    {
        float d0 = (r0 == 0) ? 0.0f : c0;
        float d1 = c1;
        float A = d0 + d1, B = NEGINF, Aex, Bex;
        block_scan_pair(A, B, Aex, Bex, sA, sB);
        float v0 = Aex + d0;
        float v1 = v0 + d1;
        int irow = rev ? (TT - 1) : 0;
        mat[irow * VV + j0] = v0;
        mat[irow * VV + j1] = v1;
        prevLds[r0 + 1] = v0;
        prevLds[r1 + 1] = v1;
        if (t == 0) prevLds[0] = NEGINF;
        __syncthreads();
    }

    // ----- 2047 remaining rows -----
    for (int s = 1; s < TT; ++s) {
        int i = rev ? (TT - 1 - s) : s;
        if (t < 100) wsubLds[t] = w[201 + 100 * ar[i] + t];   // stage sub-row
        float wd = roww[i];
        __syncthreads();

        float x0 = lse2(wd + prevLds[r0 + 1], wsubLds[e0] + prevLds[r0]);
        float x1 = lse2(wd + prevLds[r1 + 1], wsubLds[e1] + prevLds[r1]);

        float A = c0 + c1;
        float B = lse2(x0 + c1, x1);
        float Aex, Bex;
        block_scan_pair(A, B, Aex, Bex, sA, sB);

        float a0 = lse2(x0, Bex + c0);
        float a1 = lse2(x1, a0 + c1);

        mat[i * VV + j0] = a0;
        mat[i * VV + j1] = a1;
        prevLds[r0 + 1] = a0;
        prevLds[r1 + 1] = a1;
        __syncthreads();
    }
}

// ---------------- K2: global max of shifted alpha+beta ----------------
__global__ void k2_max(const float* alpha, const float* beta, unsigned* mEnc) {
    __shared__ float red[256];
    int tid = threadIdx.x;
    int gid = blockIdx.x * blockDim.x + tid;
    int stride = gridDim.x * blockDim.x;
    float lm = NEGINF;
    for (int idx = gid; idx < TT * VV; idx += stride) {
        int i = idx >> 11, j = idx & (VV - 1);
        float a = alpha[idx];
        float b = NEGINF;
        if (i < TT - 1) b = beta[idx + VV];
        if (j < VV - 1) b = fmaxf(b, beta[idx + 1]);
        if (i < TT - 1 && j < VV - 1) b = fmaxf(b, beta[idx + VV + 1]);
        lm = fmaxf(lm, a + b);
    }
    red[tid] = lm;
    __syncthreads();
    for (int s = 128; s > 0; s >>= 1) {
        if (tid < s) red[tid] = fmaxf(red[tid], red[tid + s]);
        __syncthreads();
    }
    if (tid == 0) atomicMax(mEnc, encF(red[0]));
}

// ---------------- K3: deletion (row-keyed) + insertion (col-keyed) counts ----------------
__global__ void k3_dc_ic(const float* alpha, const float* beta, const int* ar, const int* en,
                         float* sums, const unsigned* mEnc) {
    __shared__ float insBins[100];
    __shared__ float red[256];
    const int i = blockIdx.x;
    const int tid = threadIdx.x;
    float m = decF(*mEnc);
    if (!(m > -1e38f)) m = 0.0f;
    if (tid < 100) insBins[tid] = 0.0f;
    __syncthreads();
    float dcs = 0.0f;
    const float* arow = alpha + i * VV;
    for (int j = tid; j < VV; j += 256) {
        float a = arow[j];
        if (i < TT - 1) dcs += __expf(a + beta[(i + 1) * VV + j] - m);
        if (j < VV - 1) {
            float v = __expf(a + beta[i * VV + j + 1] - m);
            atomicAdd(&insBins[en[j + 1]], v);
        }
    }
    red[tid] = dcs;
    __syncthreads();
    for (int s = 128; s > 0; s >>= 1) {
        if (tid < s) red[tid] += red[tid + s];
        __syncthreads();
    }
    if (tid == 0 && i < TT - 1) atomicAdd(&sums[1 + ar[i + 1]], red[0]);
    __syncthreads();
    if (tid < 100 && insBins[tid] != 0.0f) atomicAdd(&sums[101 + tid], insBins[tid]);
}

// ---------------- K4: Q = P * B_onehot via V_WMMA_F32_16X16X4_F32 ----------------
// One wave per block; block b owns P rows [16b, 16b+16). Tiles of alpha/beta staged
// to LDS with async-to-LDS; 112 e-columns = 7 accumulator tiles.
__global__ void __launch_bounds__(32)
k4_subgemm(const float* alpha, const float* beta, const int* en,
           const unsigned* mEnc, float* q) {
    __shared__ float aRaw[256];
    __shared__ float bRaw[256];
    __shared__ float pl[256];
    __shared__ int   el[16];

    const int lane = threadIdx.x;
    const int i0 = blockIdx.x * 16;
    float m = decF(*mEnc);
    if (!(m > -1e38f)) m = 0.0f;

    v8f acc[7];
#pragma unroll
    for (int et = 0; et < 7; ++et) acc[et] = 0.0f;

    const unsigned aLdsBase = (unsigned)(uintptr_t)(&aRaw[0]);
    const unsigned bLdsBase = (unsigned)(uintptr_t)(&bRaw[0]);

    for (int jb = 0; jb < VV - 1; jb += 16) {
        // stage alpha[i0..i0+15][jb..jb+15] and beta[+1][+1] tiles (clamped addresses)
#pragma unroll
        for (int p = 0; p < 8; ++p) {
            int linear = p * 32 + lane;
            int r = linear >> 4, cjj = linear & 15;
            int ia = i0 + r;        if (ia > TT - 1) ia = TT - 1;
            int ja = jb + cjj;      if (ja > VV - 1) ja = VV - 1;
            int ib = i0 + r + 1;    if (ib > TT - 1) ib = TT - 1;
            int jc = jb + cjj + 1;  if (jc > VV - 1) jc = VV - 1;
            async_load_f32(aLdsBase + (unsigned)(linear * 4), alpha, (unsigned)((ia * VV + ja) * 4));
            async_load_f32(bLdsBase + (unsigned)(linear * 4), beta,  (unsigned)((ib * VV + jc) * 4));
        }
        wait_async0();
        __syncthreads();

        // P tile = exp(alpha + beta_shift - m), zero outside the valid domain
#pragma unroll
        for (int p = 0; p < 8; ++p) {
            int linear = p * 32 + lane;
            int r = linear >> 4, cjj = linear & 15;
            bool valid = (i0 + r < TT - 1) && (jb + cjj < VV - 1);
            pl[linear] = valid ? __expf(aRaw[linear] + bRaw[linear] - m) : 0.0f;
        }
        if (lane < 16) {
            int jj = jb + lane;
            el[lane] = (jj < VV - 1) ? en[jj + 1] : -1;
        }
        __syncthreads();

        const int Mn  = lane & 15;            // A: row M / B: col N
        const int hi2 = (lane >> 4) << 1;     // K offset 0 or 2 per lane half
#pragma unroll
        for (int k4 = 0; k4 < 16; k4 += 4) {
            int kb = k4 + hi2;
            v2f aOp;
            aOp.x = pl[Mn * 16 + kb];
            aOp.y = pl[Mn * 16 + kb + 1];
            int ek0 = el[kb], ek1 = el[kb + 1];
#pragma unroll
            for (int et = 0; et < 7; ++et) {
                int ebase = et << 4;
                v2f bOp;
                bOp.x = (ek0 == ebase + Mn) ? 1.0f : 0.0f;
                bOp.y = (ek1 == ebase + Mn) ? 1.0f : 0.0f;
                acc[et] = __builtin_amdgcn_wmma_f32_16x16x4_f32(
                    false, aOp, false, bOp, (short)0, acc[et], false, false);
            }
        }
        __syncthreads();
    }

    // C/D layout: VGPR r holds M = r (lanes 0-15) or r+8 (lanes 16-31); N = lane&15
    const int N = lane & 15;
    const int mAdd = (lane >> 4) << 3;
#pragma unroll
    for (int et = 0; et < 7; ++et) {
        int ebase = et << 4;
#pragma unroll
        for (int r = 0; r < 8; ++r) {
            int gi = i0 + r + mAdd;
            if (gi < TT - 1) q[gi * 112 + ebase + N] = acc[et][r];
        }
    }
}

// ---------------- K5: scatter Q rows into substitution bins ----------------
__global__ void k5_scatter(const float* q, const int* ar, float* sums) {
    int idx = blockIdx.x * blockDim.x + threadIdx.x;
    if (idx >= (TT - 1) * 100) return;
    int i = idx / 100;
    int e = idx - i * 100;
    float v = q[i * 112 + e];
    if (v > 0.0f) atomicAdd(&sums[201 + 100 * ar[i + 1] + e], v);
}

// ---------------- K6: finalize out[k] = w[k] + log(sums[k]) + m ----------------
__global__ void k6_final(const float* w, const float* sums, const unsigned* mEnc, float* out) {
    int k = blockIdx.x * blockDim.x + threadIdx.x;
    if (k >= NCC) return;
    float m = decF(*mEnc);
    if (!(m > -1e38f)) m = 0.0f;
    out[k] = (k == 0) ? NEGINF : (w[k] + __logf(sums[k]) + m);
}

// ---------------- launch ----------------
extern "C" void kernel_launch(void* const* d_in, const int* in_sizes, int n_in,
                              void* d_out, int out_size, void* d_ws, size_t ws_size,
                              hipStream_t stream) {
    const int*   ar = (const int*)d_in[0];     // (2048,)
    const int*   en = (const int*)d_in[1];     // (2048,)
    const float* w  = (const float*)d_in[2];   // (10201,)
    float* out = (float*)d_out;

    float* wsf = (float*)d_ws;
    const size_t ALPHA_OFF = 0;
    const size_t BETA_OFF  = (size_t)TT * VV;
    const size_t Q_OFF     = 2 * (size_t)TT * VV;
    const size_t SUMS_OFF  = Q_OFF + (size_t)TT * 112;
    const size_t ROWW_OFF  = SUMS_OFF + 10208;
    const size_t COLW_OFF  = ROWW_OFF + TT;
    const size_t MENC_OFF  = COLW_OFF + VV;

    float*    alpha = wsf + ALPHA_OFF;
    float*    beta  = wsf + BETA_OFF;
    float*    q     = wsf + Q_OFF;
    float*    sums  = wsf + SUMS_OFF;
    float*    roww  = wsf + ROWW_OFF;
    float*    colw  = wsf + COLW_OFF;
    unsigned* mEnc  = (unsigned*)(wsf + MENC_OFF);

    k0_init<<<40, 256, 0, stream>>>(ar, en, w, roww, colw, sums, mEnc);
    k1_fwd_bwd<<<2, 1024, 0, stream>>>(ar, en, w, roww, colw, alpha, beta);
    k2_max<<<1024, 256, 0, stream>>>(alpha, beta, mEnc);
    k3_dc_ic<<<TT, 256, 0, stream>>>(alpha, beta, ar, en, sums, mEnc);
    k4_subgemm<<<128, 32, 0, stream>>>(alpha, beta, en, mEnc, q);
    k5_scatter<<<801, 256, 0, stream>>>(q, ar, sums);
    k6_final<<<40, 256, 0, stream>>>(w, sums, mEnc, out);
}